// L2X_32925219291337
// MI455X (gfx1250) — compile-verified
//
#include <hip/hip_runtime.h>
#include <hip/hip_bf16.h>
#include <math.h>

// ---------------------------------------------------------------------------
// L2X forward: selector MLP -> softmax -> gumbel-softmax mask -> predictor MLP
// B=2048, D=1024, H=512, K=32, OUT=2   (all fp32)
// ---------------------------------------------------------------------------

typedef __attribute__((ext_vector_type(2))) float v2f;
typedef __attribute__((ext_vector_type(8))) float v8f;

#define B_DIM 2048
#define D_DIM 1024
#define H_DIM 512
#define K_SAMP 32
#define OUT_DIM 2

// ======================= WMMA fp32 GEMM:  C = act(A@W + b) ==================
// Block: 256 threads (8 waves). Block tile 64(M) x 64(N), K-chunk 32.
// Wave w: warpM = w&3, warpN = w>>2; each wave owns TWO 16x16 accumulators
// (cols warpN*32 and warpN*32+16) sharing A fragments -> independent WMMA
// chains and 2x A reuse from LDS.
// A and B are both stored K-contiguous in LDS (B transposed) with even row
// stride 34, so every fragment pair is one aligned ds_load_b64 straight into
// an adjacent VGPR pair — no glue movs before the WMMAs.
// Software pipelined: global loads for chunk k+1 in flight during compute.
#define TM 64
#define TN 64
#define BK 32
#define KS (BK + 2)   // even LDS row stride -> 8B-aligned b64 fragment loads

__global__ __launch_bounds__(256)
void gemm_bias_act(const float* __restrict__ A,   // [M,K] row-major
                   const float* __restrict__ W,   // [K,N] row-major
                   const float* __restrict__ bias,// [N]
                   float* __restrict__ C,         // [M,N]
                   int M, int K, int N, int relu)
{
    __shared__ float lds_a [TM][KS];   // [m][k]
    __shared__ float lds_bt[TN][KS];   // [n][k]  (transposed W tile)

    const int t     = threadIdx.x;
    const int lane  = t & 31;
    const int wave  = t >> 5;           // 0..7
    const int warpM = wave & 3;         // 0..3
    const int warpN = wave >> 2;        // 0..1
    const int blockM = blockIdx.x * TM;
    const int blockN = blockIdx.y * TN;

    v8f acc0 = {};
    v8f acc1 = {};

    // cooperative-load coordinates
    const int lr = t >> 3;              // A: 0..31 (rows lr, lr+32)
    const int lc = (t & 7) * 4;         // A: cols 0,4,...,28
    const int br = t >> 4;              // B: 0..15 (rows br, br+16)
    const int bc = (t & 15) * 4;        // B: cols 0,4,...,60

    // fragment coordinates (ISA 7.12.2): lanes 0-15 K{0,1}, lanes 16-31 K{2,3}
    const int frow  = lane & 15;
    const int koff  = (lane >> 4) << 1;
    const float* fa  = &lds_a [warpM * 16 + frow][0];
    const float* fb0 = &lds_bt[warpN * 32 + frow][0];
    const float* fb1 = &lds_bt[warpN * 32 + frow + 16][0];

    const float* aRow0 = A + (size_t)(blockM + lr)      * K + lc;
    const float* aRow1 = A + (size_t)(blockM + lr + 32) * K + lc;
    const float* wRow0 = W + (size_t)br        * N + blockN + bc;
    const float* wRow1 = W + (size_t)(br + 16) * N + blockN + bc;

    // --- prologue: load chunk 0 into registers, stage to LDS ---
    float4 ra0 = *(const float4*)(aRow0);
    float4 ra1 = *(const float4*)(aRow1);
    float4 rb0 = *(const float4*)(wRow0);
    float4 rb1 = *(const float4*)(wRow1);
    {
        *(v2f*)&lds_a[lr     ][lc    ] = (v2f){ra0.x, ra0.y};
        *(v2f*)&lds_a[lr     ][lc + 2] = (v2f){ra0.z, ra0.w};
        *(v2f*)&lds_a[lr + 32][lc    ] = (v2f){ra1.x, ra1.y};
        *(v2f*)&lds_a[lr + 32][lc + 2] = (v2f){ra1.z, ra1.w};
        lds_bt[bc + 0][br     ] = rb0.x; lds_bt[bc + 1][br     ] = rb0.y;
        lds_bt[bc + 2][br     ] = rb0.z; lds_bt[bc + 3][br     ] = rb0.w;
        lds_bt[bc + 0][br + 16] = rb1.x; lds_bt[bc + 1][br + 16] = rb1.y;
        lds_bt[bc + 2][br + 16] = rb1.z; lds_bt[bc + 3][br + 16] = rb1.w;
    }
    __syncthreads();

    for (int k0 = 0; k0 < K; k0 += BK) {
        const bool last = (k0 + BK >= K);

        // warm GL2 for chunk-after-next (global_prefetch_b8)
        if (k0 + 2 * BK < K) {
            __builtin_prefetch(aRow0 + k0 + 2 * BK, 0, 0);
            __builtin_prefetch(W + (size_t)(k0 + 2 * BK + br) * N + blockN + bc, 0, 0);
        }
        // issue global loads for next chunk (latency hidden under WMMAs)
        if (!last) {
            ra0 = *(const float4*)(aRow0 + k0 + BK);
            ra1 = *(const float4*)(aRow1 + k0 + BK);
            rb0 = *(const float4*)(wRow0 + (size_t)(k0 + BK) * N);
            rb1 = *(const float4*)(wRow1 + (size_t)(k0 + BK) * N);
        }

        // fragment loads: aligned b64 pairs, then WMMA burst
        v2f af[8], bf0[8], bf1[8];
        #pragma unroll
        for (int i = 0; i < 8; ++i) {
            const int kk = i * 4 + koff;
            af[i]  = *(const v2f*)(fa  + kk);
            bf0[i] = *(const v2f*)(fb0 + kk);
            bf1[i] = *(const v2f*)(fb1 + kk);
        }
        #pragma unroll
        for (int i = 0; i < 8; ++i) {
            acc0 = __builtin_amdgcn_wmma_f32_16x16x4_f32(
                       false, af[i], false, bf0[i], (short)0, acc0, false, false);
            acc1 = __builtin_amdgcn_wmma_f32_16x16x4_f32(
                       false, af[i], false, bf1[i], (short)0, acc1, false, false);
        }
        __syncthreads();   // everyone done reading LDS for this chunk

        if (!last) {
            *(v2f*)&lds_a[lr     ][lc    ] = (v2f){ra0.x, ra0.y};
            *(v2f*)&lds_a[lr     ][lc + 2] = (v2f){ra0.z, ra0.w};
            *(v2f*)&lds_a[lr + 32][lc    ] = (v2f){ra1.x, ra1.y};
            *(v2f*)&lds_a[lr + 32][lc + 2] = (v2f){ra1.z, ra1.w};
            lds_bt[bc + 0][br     ] = rb0.x; lds_bt[bc + 1][br     ] = rb0.y;
            lds_bt[bc + 2][br     ] = rb0.z; lds_bt[bc + 3][br     ] = rb0.w;
            lds_bt[bc + 0][br + 16] = rb1.x; lds_bt[bc + 1][br + 16] = rb1.y;
            lds_bt[bc + 2][br + 16] = rb1.z; lds_bt[bc + 3][br + 16] = rb1.w;
            __syncthreads();
        }
    }

    // Epilogue: C/D layout — VGPR i = rows (i, i+8), col = lane&15
    const int col0  = blockN + warpN * 32 + (lane & 15);
    const int col1  = col0 + 16;
    const int rbase = blockM + warpM * 16 + ((lane >> 4) << 3);
    const float bc0 = bias[col0];
    const float bc1 = bias[col1];
    #pragma unroll
    for (int i = 0; i < 8; ++i) {
        float v0 = acc0[i] + bc0;
        float v1 = acc1[i] + bc1;
        if (relu) { v0 = fmaxf(v0, 0.0f); v1 = fmaxf(v1, 0.0f); }
        C[(size_t)(rbase + i) * N + col0] = v0;
        C[(size_t)(rbase + i) * N + col1] = v1;
    }
}

// ======================= row softmax over D=1024 (in place) =================
__global__ __launch_bounds__(256)
void softmax_rows(float* __restrict__ logits)   // [B, D]
{
    __shared__ float red[256];
    const int t = threadIdx.x;
    const size_t base = (size_t)blockIdx.x * D_DIM + t * 4;

    float4 v = *(const float4*)(logits + base);
    float lmax = fmaxf(fmaxf(v.x, v.y), fmaxf(v.z, v.w));
    red[t] = lmax; __syncthreads();
    for (int s = 128; s > 0; s >>= 1) {
        if (t < s) red[t] = fmaxf(red[t], red[t + s]);
        __syncthreads();
    }
    const float rmax = red[0]; __syncthreads();

    v.x = __expf(v.x - rmax); v.y = __expf(v.y - rmax);
    v.z = __expf(v.z - rmax); v.w = __expf(v.w - rmax);
    red[t] = v.x + v.y + v.z + v.w; __syncthreads();
    for (int s = 128; s > 0; s >>= 1) {
        if (t < s) red[t] += red[t + s];
        __syncthreads();
    }
    const float inv = 1.0f / red[0];
    v.x *= inv; v.y *= inv; v.z *= inv; v.w *= inv;
    *(float4*)(logits + base) = v;
}

// ======= gumbel-softmax mask: single pass over uniform[B,K,D] (256 MB) ======
// One block per batch row b. masks[b,d] = max_k softmax_d(gumbel + logit),
// new_inputs[b,d] = x[b,d] * masks[b,d].
__global__ __launch_bounds__(256)
void gumbel_mask(const float* __restrict__ x,        // [B, D]
                 const float* __restrict__ uniform,  // [B, K, D]
                 const float* __restrict__ logits,   // [B, D] (softmaxed)
                 float* __restrict__ masks,          // [B, D]
                 float* __restrict__ new_inputs)     // [B, D]
{
    __shared__ float red[256];
    const int t = threadIdx.x;
    const int b = blockIdx.x;
    const size_t rowbase = (size_t)b * D_DIM + t * 4;

    const float4 lg4 = *(const float4*)(logits + rowbase);
    const float lg[4] = { lg4.x, lg4.y, lg4.z, lg4.w };
    float m[4] = { 0.0f, 0.0f, 0.0f, 0.0f };

    for (int k = 0; k < K_SAMP; ++k) {
        const size_t ubase = (((size_t)b * K_SAMP + k) * D_DIM) + t * 4;
        const float4 u4 = *(const float4*)(uniform + ubase);
        float nz[4];
        // gumbel = -log(-log(u)); TEMPERATURE == 1
        nz[0] = -logf(-logf(u4.x)) + lg[0];
        nz[1] = -logf(-logf(u4.y)) + lg[1];
        nz[2] = -logf(-logf(u4.z)) + lg[2];
        nz[3] = -logf(-logf(u4.w)) + lg[3];

        float lmax = fmaxf(fmaxf(nz[0], nz[1]), fmaxf(nz[2], nz[3]));
        red[t] = lmax; __syncthreads();
        for (int s = 128; s > 0; s >>= 1) {
            if (t < s) red[t] = fmaxf(red[t], red[t + s]);
            __syncthreads();
        }
        const float rmax = red[0]; __syncthreads();

        float lsum = 0.0f;
        #pragma unroll
        for (int j = 0; j < 4; ++j) { nz[j] = __expf(nz[j] - rmax); lsum += nz[j]; }
        red[t] = lsum; __syncthreads();
        for (int s = 128; s > 0; s >>= 1) {
            if (t < s) red[t] += red[t + s];
            __syncthreads();
        }
        const float inv = 1.0f / red[0]; __syncthreads();

        #pragma unroll
        for (int j = 0; j < 4; ++j) m[j] = fmaxf(m[j], nz[j] * inv);
    }

    const float4 x4 = *(const float4*)(x + rowbase);
    float4 m4  = { m[0], m[1], m[2], m[3] };
    float4 ni4 = { x4.x * m[0], x4.y * m[1], x4.z * m[2], x4.w * m[3] };
    *(float4*)(masks + rowbase)      = m4;
    *(float4*)(new_inputs + rowbase) = ni4;
}

// ================= final projection: deep_out = hp @ wp1 + bp1 (N=2) ========
__global__ __launch_bounds__(256)
void final_proj(const float* __restrict__ hp,   // [B, H]
                const float* __restrict__ wp1,  // [H, 2]
                const float* __restrict__ bp1,  // [2]
                float* __restrict__ out)        // [B, 2]
{
    __shared__ float r0[256];
    __shared__ float r1[256];
    const int t = threadIdx.x;
    const int b = blockIdx.x;
    const float h0 = hp[(size_t)b * H_DIM + t];
    const float h1 = hp[(size_t)b * H_DIM + t + 256];
    r0[t] = h0 * wp1[t * 2 + 0] + h1 * wp1[(t + 256) * 2 + 0];
    r1[t] = h0 * wp1[t * 2 + 1] + h1 * wp1[(t + 256) * 2 + 1];
    __syncthreads();
    for (int s = 128; s > 0; s >>= 1) {
        if (t < s) { r0[t] += r0[t + s]; r1[t] += r1[t + s]; }
        __syncthreads();
    }
    if (t == 0) {
        out[(size_t)b * 2 + 0] = r0[0] + bp1[0];
        out[(size_t)b * 2 + 1] = r1[0] + bp1[1];
    }
}

// ===========================================================================
extern "C" void kernel_launch(void* const* d_in, const int* in_sizes, int n_in,
                              void* d_out, int out_size, void* d_ws, size_t ws_size,
                              hipStream_t stream)
{
    const float* x       = (const float*)d_in[0];   // [B, D]
    const float* uniform = (const float*)d_in[1];   // [B, K, D]
    const float* w0      = (const float*)d_in[2];   // [D, H]
    const float* b0      = (const float*)d_in[3];   // [H]
    const float* w1      = (const float*)d_in[4];   // [H, H]
    const float* b1      = (const float*)d_in[5];   // [H]
    const float* wout    = (const float*)d_in[6];   // [H, D]
    const float* bout    = (const float*)d_in[7];   // [D]
    const float* wp0     = (const float*)d_in[8];   // [D, H]
    const float* bp0     = (const float*)d_in[9];   // [H]
    const float* wp1     = (const float*)d_in[10];  // [H, OUT]
    const float* bp1     = (const float*)d_in[11];  // [OUT]
    (void)in_sizes; (void)n_in; (void)out_size; (void)ws_size;

    // Workspace layout (floats): h1 | h2 | logits | new_inputs | hp  (~28 MB)
    float* ws      = (float*)d_ws;
    float* h1      = ws;                                  // [B, H]
    float* h2      = h1 + (size_t)B_DIM * H_DIM;          // [B, H]
    float* logits  = h2 + (size_t)B_DIM * H_DIM;          // [B, D]
    float* newin   = logits + (size_t)B_DIM * D_DIM;      // [B, D]
    float* hp      = newin + (size_t)B_DIM * D_DIM;       // [B, H]

    // Output layout: deep_out [B,2] first, then masks [B,D]
    float* deep_out = (float*)d_out;
    float* masks    = deep_out + (size_t)B_DIM * OUT_DIM;

    dim3 blk(256);

    // 1) h1 = relu(x @ w0 + b0)            [2048,1024]x[1024,512]
    gemm_bias_act<<<dim3(B_DIM / TM, H_DIM / TN), blk, 0, stream>>>(
        x, w0, b0, h1, B_DIM, D_DIM, H_DIM, 1);
    // 2) h2 = relu(h1 @ w1 + b1)           [2048,512]x[512,512]
    gemm_bias_act<<<dim3(B_DIM / TM, H_DIM / TN), blk, 0, stream>>>(
        h1, w1, b1, h2, B_DIM, H_DIM, H_DIM, 1);
    // 3) logits = h2 @ wout + bout         [2048,512]x[512,1024]
    gemm_bias_act<<<dim3(B_DIM / TM, D_DIM / TN), blk, 0, stream>>>(
        h2, wout, bout, logits, B_DIM, H_DIM, D_DIM, 0);
    // 4) logits = softmax(logits, axis=-1)
    softmax_rows<<<dim3(B_DIM), blk, 0, stream>>>(logits);
    // 5) masks + new_inputs (single pass over uniform)
    gumbel_mask<<<dim3(B_DIM), blk, 0, stream>>>(x, uniform, logits, masks, newin);
    // 6) hp = relu(new_inputs @ wp0 + bp0) [2048,1024]x[1024,512]
    gemm_bias_act<<<dim3(B_DIM / TM, H_DIM / TN), blk, 0, stream>>>(
        newin, wp0, bp0, hp, B_DIM, D_DIM, H_DIM, 1);
    // 7) deep_out = hp @ wp1 + bp1         [2048,512]x[512,2]
    final_proj<<<dim3(B_DIM), blk, 0, stream>>>(hp, wp1, bp1, deep_out);
}